// Generator_55808805044519
// MI455X (gfx1250) — compile-verified
//
#include <hip/hip_runtime.h>
#include <math.h>

#define BB  64
#define HH  128
#define TT  12
#define MM  50000
#define KK  3
#define EE  32
#define NDD 128
#define NCC 4
#define NN  150000   /* K*M; id NN is the end token */

/* workspace layout (float offsets) */
#define OFF_H      0        /* 8192  */
#define OFF_C      8192     /* 8192  */
#define OFF_X      16384    /* 8192  */
#define OFF_GATES  24576    /* 32768 */
#define OFF_PROJ   57344    /* 6144  */
#define OFF_PROJSQ 63488    /* 192   */
#define OFF_BEST   63744    /* 64 x u64 (8B aligned) */
#define OFF_TIDX   63872    /* 64 int */
#define OFF_KSEL   63936    /* 64 int */

typedef float v2f __attribute__((ext_vector_type(2)));
typedef float v8f __attribute__((ext_vector_type(8)));

__device__ __forceinline__ float sigf(float x) { return 1.0f / (1.0f + expf(-x)); }

/* ---------------- zero the one-hot node output (460 MB) ---------------- */
__global__ void k_zero(float4* p, long long n4) {
    long long i = (long long)blockIdx.x * blockDim.x + threadIdx.x;
    if (i < n4) p[i] = make_float4(0.f, 0.f, 0.f, 0.f);
}

/* ---------------- init: inter = tanh(z W1^T + b1); h0,c0 --------------- */
__global__ void __launch_bounds__(256) k_init(const float* z, const float* W1, const float* b1,
                                              const float* Wc, const float* bc,
                                              const float* Wh, const float* bh, float* ws) {
    __shared__ float s_inter[BB * HH];
    int tid = threadIdx.x;
    for (int idx = tid; idx < BB * HH; idx += 256) {
        int b = idx / HH, j = idx % HH;
        float acc = b1[j];
        const float* zr = z + b * NDD;
        const float* wr = W1 + j * NDD;
        for (int k = 0; k < NDD; k++) acc += zr[k] * wr[k];
        s_inter[idx] = tanhf(acc);
    }
    __syncthreads();
    for (int idx = tid; idx < BB * HH; idx += 256) {
        int b = idx / HH, j = idx % HH;
        const float* ir  = s_inter + b * HH;
        const float* wcr = Wc + j * HH;
        const float* whr = Wh + j * HH;
        float ah = bc[j], ac = bh[j];
        for (int k = 0; k < HH; k++) { ah += ir[k] * wcr[k]; ac += ir[k] * whr[k]; }
        ws[OFF_H + idx] = tanhf(ah);
        ws[OFF_C + idx] = tanhf(ac);
        ws[OFF_X + idx] = 0.0f;
    }
}

/* ------------- gates = x Wih^T + h Whh^T + bih + bhh  (WMMA f32) ------- */
__global__ void __launch_bounds__(256) k_gates(const float* Wih, const float* bih,
                                               const float* Whh, const float* bhh, float* ws) {
    const int lane = threadIdx.x & 31;
    const int wv   = threadIdx.x >> 5;
    const int lmod = lane & 15;
    const int lhi  = lane >> 4;
    const int koff = 2 * lhi;
    const float* x = ws + OFF_X;
    const float* h = ws + OFF_H;
    float* g = ws + OFF_GATES;
    for (int i = 0; i < 16; i++) {
        int tile = wv + 8 * i;                 /* 128 tiles: 4 b-tiles x 32 j-tiles */
        int jt = tile & 31, bt = tile >> 5;
        int j0 = jt * 16, b0 = bt * 16;
        v8f acc = {0.f, 0.f, 0.f, 0.f, 0.f, 0.f, 0.f, 0.f};
        const float* arowx = x   + (b0 + lmod) * HH + koff;
        const float* arowh = h   + (b0 + lmod) * HH + koff;
        const float* browi = Wih + (j0 + lmod) * HH + koff;
        const float* browh = Whh + (j0 + lmod) * HH + koff;
        for (int k0 = 0; k0 < HH; k0 += 4) {
            v2f a = *(const v2f*)(arowx + k0);
            v2f b = *(const v2f*)(browi + k0);
            acc = __builtin_amdgcn_wmma_f32_16x16x4_f32(false, a, false, b, (short)0, acc, false, false);
        }
        for (int k0 = 0; k0 < HH; k0 += 4) {
            v2f a = *(const v2f*)(arowh + k0);
            v2f b = *(const v2f*)(browh + k0);
            acc = __builtin_amdgcn_wmma_f32_16x16x4_f32(false, a, false, b, (short)0, acc, false, false);
        }
        int j = j0 + lmod;
        float bias = bih[j] + bhh[j];
#pragma unroll
        for (int v = 0; v < 8; v++) {
            int b = b0 + v + 8 * lhi;          /* C: M = v + 8*(lane/16), N = lane%16 */
            g[b * (4 * HH) + j] = acc[v] + bias;
        }
    }
}

/* ---- LSTM elementwise, type argmax (hard one-hot), proj, best reset --- */
__global__ void __launch_bounds__(256) k_cell(const float* Wt, const float* bt_,
                                              const float* gt, const float* Wn, const float* bn,
                                              float* ws, float* out_type, int t) {
    __shared__ float sh[BB * HH];
    int tid = threadIdx.x;
    float* hS = ws + OFF_H;
    float* cS = ws + OFF_C;
    const float* g = ws + OFF_GATES;
    for (int idx = tid; idx < BB * HH; idx += 256) {
        int b = idx / HH, j = idx % HH;
        const float* gr = g + b * 4 * HH;
        float gi = gr[j], gf = gr[HH + j], gg = gr[2 * HH + j], go = gr[3 * HH + j];
        float cn = sigf(gf) * cS[idx] + sigf(gi) * tanhf(gg);
        float hn = sigf(go) * tanhf(cn);
        cS[idx] = cn; hS[idx] = hn; sh[idx] = hn;
    }
    __syncthreads();
    int* tidx = (int*)(ws + OFF_TIDX);
    int* ksel = (int*)(ws + OFF_KSEL);
    unsigned long long* best = (unsigned long long*)(ws + OFF_BEST);
    if (tid < BB) {
        int b = tid;
        const float* hr = sh + b * HH;
        float lg[NCC];
        for (int c = 0; c < NCC; c++) {
            float acc = bt_[c] + gt[(size_t)t * BB * NCC + b * NCC + c];
            const float* wr = Wt + c * HH;
            for (int k = 0; k < HH; k++) acc += hr[k] * wr[k];
            lg[c] = acc;
        }
        int am = 0; float bv = lg[0];
        for (int c = 1; c < NCC; c++) if (lg[c] > bv) { bv = lg[c]; am = c; }
        tidx[b] = am;
        ksel[b] = (am < KK) ? am : (KK - 1);
        best[b] = 0ull;
        for (int c = 0; c < NCC; c++)
            out_type[b * TT * NCC + t * NCC + c] = (c == am) ? 1.0f : 0.0f;
    }
    float* proj = ws + OFF_PROJ;
    for (int idx = tid; idx < KK * BB * EE; idx += 256) {
        int k = idx / (BB * EE); int r = idx % (BB * EE);
        int b = r / EE, e = r % EE;
        float acc = bn[k * EE + e];
        const float* hr = sh + b * HH;
        const float* wr = Wn + (k * EE + e) * HH;
        for (int q = 0; q < HH; q++) acc += hr[q] * wr[q];
        proj[(k * BB + b) * EE + e] = acc;
    }
    __syncthreads();
    float* projsq = ws + OFF_PROJSQ;
    if (tid < KK * BB) {
        const float* pr = ws + OFF_PROJ + tid * EE;
        float s = 0.f;
        for (int e = 0; e < EE; e++) s += pr[e] * pr[e];
        projsq[tid] = s;
    }
}

/* ----- distance GEMM (WMMA f32) + gumbel-max argmax over M=50000 ------- */
__global__ void __launch_bounds__(256) k_dist(const float* emb, const float* gn,
                                              float* ws, int t) {
    __shared__ float s_proj[BB * EE];
    __shared__ float s_psq[BB];
    __shared__ int   s_ksel[BB];
    const int k = blockIdx.y;
    int tid = threadIdx.x;
    const float* proj = ws + OFF_PROJ + (size_t)k * BB * EE;
    for (int i = tid; i < BB * EE; i += 256) s_proj[i] = proj[i];
    if (tid < BB) {
        s_psq[tid]  = ws[OFF_PROJSQ + k * BB + tid];
        s_ksel[tid] = ((int*)(ws + OFF_KSEL))[tid];
    }
    __syncthreads();
    const int lane = tid & 31, wv = tid >> 5;
    int mt = blockIdx.x * 8 + wv;               /* 3125 m-tiles of 16 */
    if (mt >= MM / 16) return;                  /* wave-uniform exit */
    const int m0 = mt * 16;
    const int lmod = lane & 15, lhi = lane >> 4, koff = 2 * lhi;
    const float* erow = emb + ((size_t)k * MM + m0 + lmod) * EE + koff;
    v2f bfr[8];
    float esq = 0.f;
#pragma unroll
    for (int e = 0; e < 8; e++) {
        bfr[e] = *(const v2f*)(erow + 4 * e);
        esq += bfr[e][0] * bfr[e][0] + bfr[e][1] * bfr[e][1];
    }
    esq += __shfl_xor(esq, 16, 32);             /* lanes L and L^16 share m */
    const int m = m0 + lmod;
    unsigned long long* best = (unsigned long long*)(ws + OFF_BEST);
    for (int bt = 0; bt < 4; bt++) {
        v8f acc = {0.f, 0.f, 0.f, 0.f, 0.f, 0.f, 0.f, 0.f};
        const float* arow = s_proj + (bt * 16 + lmod) * EE + koff;
#pragma unroll
        for (int e = 0; e < 8; e++) {
            v2f a = *(const v2f*)(arow + 4 * e);
            acc = __builtin_amdgcn_wmma_f32_16x16x4_f32(false, a, false, bfr[e], (short)0, acc, false, false);
        }
#pragma unroll
        for (int v = 0; v < 8; v++) {
            int b = bt * 16 + v + 8 * lhi;      /* half-wave-uniform branch below */
            if (s_ksel[b] == k) {
                float d2   = esq + s_psq[b] - 2.0f * acc[v];
                float dist = sqrtf(fmaxf(d2, 0.0f));
                float sc   = gn[(size_t)t * BB * MM + (size_t)b * MM + m] - dist;
                unsigned int fk = __float_as_uint(sc);
                fk = (fk & 0x80000000u) ? ~fk : (fk | 0x80000000u);
                unsigned long long pk =
                    ((unsigned long long)fk << 32) | (unsigned long long)(0xFFFFFFFFu - (unsigned int)m);
#pragma unroll
                for (int d = 1; d < 16; d <<= 1) {  /* reduce within each half-wave */
                    unsigned int phi = __shfl_xor((unsigned int)(pk >> 32), d, 32);
                    unsigned int plo = __shfl_xor((unsigned int)(pk & 0xFFFFFFFFu), d, 32);
                    unsigned long long other = ((unsigned long long)phi << 32) | plo;
                    if (other > pk) pk = other;
                }
                if (lmod == 0) atomicMax(best + b, pk);
            }
        }
    }
}

/* --------- node select, one-hot scatter, x_new via column gathers ------ */
__global__ void __launch_bounds__(256) k_select(const int* type_ids, const float* Wdt,
                                                const float* bdt, const float* Wdn,
                                                const float* bdn, float* ws,
                                                float* out_node, int t) {
    __shared__ int s_node[BB], s_t[BB];
    int tid = threadIdx.x;
    if (tid < BB) {
        unsigned long long pk = ((unsigned long long*)(ws + OFF_BEST))[tid];
        int m   = (int)(0xFFFFFFFFu - (unsigned int)(pk & 0xFFFFFFFFu));
        int tix = ((int*)(ws + OFF_TIDX))[tid];
        int k   = (tix < KK) ? tix : (KK - 1);
        int node = (tix == KK) ? NN : type_ids[k * MM + m];
        s_node[tid] = node; s_t[tid] = tix;
        out_node[(size_t)(tid * TT + t) * (NN + 1) + node] = 1.0f;
    }
    __syncthreads();
    float* x = ws + OFF_X;
    for (int idx = tid; idx < BB * HH; idx += 256) {
        int b = idx / HH, j = idx % HH;
        x[idx] = Wdt[j * NCC + s_t[b]] + bdt[j]
               + Wdn[(size_t)j * (NN + 1) + s_node[b]] + bdn[j];
    }
}

extern "C" void kernel_launch(void* const* d_in, const int* in_sizes, int n_in,
                              void* d_out, int out_size, void* d_ws, size_t ws_size,
                              hipStream_t stream) {
    (void)in_sizes; (void)n_in; (void)out_size; (void)ws_size;
    const float* z   = (const float*)d_in[0];
    const float* gt  = (const float*)d_in[1];
    const float* gn  = (const float*)d_in[2];
    const float* W1  = (const float*)d_in[3];
    const float* b1  = (const float*)d_in[4];
    const float* Wc  = (const float*)d_in[5];
    const float* bc  = (const float*)d_in[6];
    const float* Wh  = (const float*)d_in[7];
    const float* bh  = (const float*)d_in[8];
    const float* Wih = (const float*)d_in[9];
    const float* bih = (const float*)d_in[10];
    const float* Whh = (const float*)d_in[11];
    const float* bhh = (const float*)d_in[12];
    const float* Wt  = (const float*)d_in[13];
    const float* btt = (const float*)d_in[14];
    const float* Wn  = (const float*)d_in[15];
    const float* bn  = (const float*)d_in[16];
    const float* Wdt = (const float*)d_in[17];
    const float* bdt = (const float*)d_in[18];
    const float* Wdn = (const float*)d_in[19];
    const float* bdn = (const float*)d_in[20];
    const float* emb = (const float*)d_in[21];
    const int*   tds = (const int*)d_in[22];

    float* out      = (float*)d_out;
    float* out_type = out;
    float* out_node = out + BB * TT * NCC;
    float* ws       = (float*)d_ws;

    long long n4 = (long long)BB * TT * (NN + 1) / 4;
    k_zero<<<dim3((unsigned)((n4 + 255) / 256)), 256, 0, stream>>>((float4*)out_node, n4);
    k_init<<<1, 256, 0, stream>>>(z, W1, b1, Wc, bc, Wh, bh, ws);
    for (int t = 0; t < TT; t++) {
        k_gates<<<1, 256, 0, stream>>>(Wih, bih, Whh, bhh, ws);
        k_cell<<<1, 256, 0, stream>>>(Wt, btt, gt, Wn, bn, ws, out_type, t);
        k_dist<<<dim3((MM / 16 + 7) / 8, KK), 256, 0, stream>>>(emb, gn, ws, t);
        k_select<<<1, 256, 0, stream>>>(tds, Wdt, bdt, Wdn, bdn, ws, out_node, t);
    }
}